// STGCNdiffAttentionLayer_85899345920218
// MI455X (gfx1250) — compile-verified
//
#include <hip/hip_runtime.h>
#include <math.h>

typedef __attribute__((ext_vector_type(2))) float v2f;
typedef __attribute__((ext_vector_type(4))) float v4f;
typedef __attribute__((ext_vector_type(8))) float v8f;
typedef __attribute__((ext_vector_type(4))) int   v4i;

#define B_   4
#define L_   64
#define S_   64
#define N_   768
#define D_   64
#define H_   8
#define DK_  8
#define CIN_ 192   // (ORDER*SUP+1)*D

// ---------------------------------------------------------------------------
// Async global->LDS staging (CDNA5 GLOBAL_LOAD_ASYNC_TO_LDS_B128, ASYNCcnt),
// gated on builtin availability with a synchronous VGPR-mediated fallback.
// Probe result (round 2): builtin exists and takes
//   (v4i addrspace(1)* gsrc, v4i addrspace(3)* ldst, imm offset, imm cpol).
// ---------------------------------------------------------------------------
#if defined(__has_builtin)
#if __has_builtin(__builtin_amdgcn_global_load_async_to_lds_b128) && \
    __has_builtin(__builtin_amdgcn_s_wait_asynccnt)
#define USE_ASYNC_LDS 1
#endif
#endif

#ifdef USE_ASYNC_LDS
typedef v4i __attribute__((address_space(1))) gv4i;  // global int4
typedef v4i __attribute__((address_space(3))) lv4i;  // LDS int4
#define WAIT_ASYNC(n) __builtin_amdgcn_s_wait_asynccnt(n)
#else
#define WAIT_ASYNC(n)
#endif

__device__ __forceinline__ void copy16_g2l(const float* __restrict__ g,
                                           float* __restrict__ l) {
#ifdef USE_ASYNC_LDS
  __builtin_amdgcn_global_load_async_to_lds_b128((gv4i*)(float*)g,
                                                 (lv4i*)l, 0, 0);
#else
  *(v4f*)l = *(const v4f*)g;
#endif
}

// ---------------------------------------------------------------------------
// Diffusion hop: per slice (b,s):  Y[w,d] = sum_v A[v,w] * X[v,d]
//   A: [N,N] row-major (diffusion matrix), X,Y: [B*S, N, D] row-major.
// One workgroup computes a 64(w) x 64(d) tile for one slice.
// 8 waves; wave (ww): rows mblk = (ww&3)*16, cols nblk = (ww>>2)*32 (2 tiles).
// fp32 WMMA 16x16x4; K staged through double-buffered LDS via async copies:
// chunk i+1 is issued before waiting on chunk i (in-order async completion,
// so s_wait_asynccnt 8 guarantees the older 8 ops per wave have landed).
// ---------------------------------------------------------------------------
__global__ __launch_bounds__(256)
void hop_kernel(const float* __restrict__ A,
                const float* __restrict__ X,
                float* __restrict__ Y) {
  __shared__ float ldsA[2][64][64];  // [buf][k = v][m = w]
  __shared__ float ldsB[2][64][64];  // [buf][k = v][n = d]

  const int w0    = blockIdx.x * 64;
  const int slice = blockIdx.y;                    // b*S + s
  const size_t sbase = (size_t)slice * (N_ * D_);

  const int t    = threadIdx.x;
  const int lane = t & 31;
  const int wave = t >> 5;
  const int mblk = (wave & 3) * 16;
  const int nblk = (wave >> 2) * 32;

  const int krow  = t >> 2;          // 0..63 : LDS row staged by this thread
  const int kcol4 = (t & 3) * 16;    // column group base (16 floats)

  v8f acc0 = {};
  v8f acc1 = {};

  const int fm = lane & 15;          // M (or N) within fragment
  const int kb = (lane >> 4) * 2;    // K pair: lanes 0-15 -> K=0,1 ; 16-31 -> K=2,3

  auto stage = [&](int buf, int v0) {
    const float* arow = A + (size_t)(v0 + krow) * N_ + w0 + kcol4;
    const float* xrow = X + sbase + (size_t)(v0 + krow) * D_ + kcol4;
#pragma unroll
    for (int j = 0; j < 4; ++j) {
      copy16_g2l(arow + 4 * j, &ldsA[buf][krow][kcol4 + 4 * j]);
      copy16_g2l(xrow + 4 * j, &ldsB[buf][krow][kcol4 + 4 * j]);
    }
  };

  stage(0, 0);
  const int NCHUNK = N_ / 64;  // 12
  for (int chunk = 0; chunk < NCHUNK; ++chunk) {
    const int cur = chunk & 1;
    if (chunk + 1 < NCHUNK) {
      stage(cur ^ 1, (chunk + 1) * 64);  // prefetch next chunk (other buffer)
      WAIT_ASYNC(8);                     // older 8 ops (current chunk) done
    } else {
      WAIT_ASYNC(0);
    }
    __syncthreads();

#pragma unroll
    for (int kk = 0; kk < 64; kk += 4) {
      v2f a, b0, b1;
      // A-frag (16x4): A^T[w, v] = A[v, w]  -> ldsA[k][m]
      a.x  = ldsA[cur][kk + kb][mblk + fm];
      a.y  = ldsA[cur][kk + kb + 1][mblk + fm];
      // B-frags (4x16): X[v, d] -> ldsB[k][n]
      b0.x = ldsB[cur][kk + kb][nblk + fm];
      b0.y = ldsB[cur][kk + kb + 1][nblk + fm];
      b1.x = ldsB[cur][kk + kb][nblk + 16 + fm];
      b1.y = ldsB[cur][kk + kb + 1][nblk + 16 + fm];
      acc0 = __builtin_amdgcn_wmma_f32_16x16x4_f32(false, a, false, b0,
                                                   (short)0, acc0, false, false);
      acc1 = __builtin_amdgcn_wmma_f32_16x16x4_f32(false, a, false, b1,
                                                   (short)0, acc1, false, false);
    }
    __syncthreads();
  }

  // C/D layout: VGPR i: lanes 0-15 -> M=i, N=lane ; lanes 16-31 -> M=i+8.
  const int rhalf = (lane >> 4) * 8;
#pragma unroll
  for (int i = 0; i < 8; ++i) {
    const int row = mblk + i + rhalf;
    Y[sbase + (size_t)(w0 + row) * D_ + nblk + fm]      = acc0[i];
    Y[sbase + (size_t)(w0 + row) * D_ + nblk + 16 + fm] = acc1[i];
  }
}

// ---------------------------------------------------------------------------
// Generic row-projection GEMM:
//   out[r, e] = sum_{c=0}^{64*nchunks-1} Xcat[r, c] * W[e, c] + bias[e]
// Xcat = horizontal concat of up to 3 row-blocks X0,X1,X2 (each [rows,64]).
// W row-major [64, ldw]. One workgroup: 64 rows x 64 outputs.
// ---------------------------------------------------------------------------
__global__ __launch_bounds__(256)
void proj_kernel(const float* __restrict__ X0, const float* __restrict__ X1,
                 const float* __restrict__ X2, int nchunks,
                 const float* __restrict__ W, int ldw,
                 const float* __restrict__ bias, float* __restrict__ out) {
  __shared__ float ldsX[64][64];  // [m = row][k = channel-in-chunk]
  __shared__ float ldsW[64][64];  // [k][e]

  const size_t row0 = (size_t)blockIdx.x * 64;

  const int t    = threadIdx.x;
  const int lane = t & 31;
  const int wave = t >> 5;
  const int mblk = (wave & 3) * 16;
  const int nblk = (wave >> 2) * 32;

  const int lrow  = t >> 2;          // 0..63
  const int lcol4 = (t & 3) * 16;

  v8f acc0 = {};
  v8f acc1 = {};

  const int fm = lane & 15;
  const int kb = (lane >> 4) * 2;

  const float* Xs[3] = {X0, X1, X2};
  for (int c = 0; c < nchunks; ++c) {
    const float* Xc = Xs[c];
    const float* xrow = Xc + (row0 + lrow) * 64 + lcol4;
    const float* wrow = W + (size_t)lrow * ldw + c * 64 + lcol4;
#pragma unroll
    for (int j = 0; j < 4; ++j) {
      // ldsX[m][k] = Xc[(row0+m)*64 + k]
      *(v4f*)&ldsX[lrow][lcol4 + 4 * j] = *(const v4f*)(xrow + 4 * j);
      // ldsW[k][e] = W[e*ldw + c*64 + k]  (transpose while staging)
      v4f wv = *(const v4f*)(wrow + 4 * j);
      ldsW[lcol4 + 4 * j + 0][lrow] = wv.x;
      ldsW[lcol4 + 4 * j + 1][lrow] = wv.y;
      ldsW[lcol4 + 4 * j + 2][lrow] = wv.z;
      ldsW[lcol4 + 4 * j + 3][lrow] = wv.w;
    }
    __syncthreads();

#pragma unroll
    for (int kk = 0; kk < 64; kk += 4) {
      v2f a, b0, b1;
      a.x  = ldsX[mblk + fm][kk + kb];
      a.y  = ldsX[mblk + fm][kk + kb + 1];
      b0.x = ldsW[kk + kb][nblk + fm];
      b0.y = ldsW[kk + kb + 1][nblk + fm];
      b1.x = ldsW[kk + kb][nblk + 16 + fm];
      b1.y = ldsW[kk + kb + 1][nblk + 16 + fm];
      acc0 = __builtin_amdgcn_wmma_f32_16x16x4_f32(false, a, false, b0,
                                                   (short)0, acc0, false, false);
      acc1 = __builtin_amdgcn_wmma_f32_16x16x4_f32(false, a, false, b1,
                                                   (short)0, acc1, false, false);
    }
    __syncthreads();
  }

  const int rhalf = (lane >> 4) * 8;
  const float be0 = bias[nblk + fm];
  const float be1 = bias[nblk + 16 + fm];
#pragma unroll
  for (int i = 0; i < 8; ++i) {
    const int row = mblk + i + rhalf;
    out[(row0 + row) * 64 + nblk + fm]      = acc0[i] + be0;
    out[(row0 + row) * 64 + nblk + 16 + fm] = acc1[i] + be1;
  }
}

// ---------------------------------------------------------------------------
// Fuse GCN mlp with the following K/V projection:
//   Wc[e,c] = sum_o Wp[e,o] * Wg[o,c]   (64 x 192)
//   bc[e]   = bp[e] + sum_o Wp[e,o] * bg[o]
// ---------------------------------------------------------------------------
__global__ __launch_bounds__(256)
void combine_w_kernel(const float* __restrict__ Wp, const float* __restrict__ bp,
                      const float* __restrict__ Wg, const float* __restrict__ bg,
                      float* __restrict__ Wc, float* __restrict__ bc) {
  for (int idx = threadIdx.x; idx < 64 * CIN_; idx += blockDim.x) {
    const int e = idx / CIN_;
    const int c = idx % CIN_;
    float s = 0.f;
#pragma unroll 8
    for (int o = 0; o < 64; ++o) s += Wp[e * 64 + o] * Wg[o * CIN_ + c];
    Wc[idx] = s;
  }
  if (threadIdx.x < 64) {
    const int e = threadIdx.x;
    float s = bp[e];
#pragma unroll 8
    for (int o = 0; o < 64; ++o) s += Wp[e * 64 + o] * bg[o];
    bc[e] = s;
  }
}

// ---------------------------------------------------------------------------
// Per-node attention over time. One workgroup per (b, n).
// q,k,v: [B, T, N, 64]. Each thread owns two (h, l) rows end-to-end:
// scores (K=8 dot), softmax, AV (S=64) all in registers.
// ---------------------------------------------------------------------------
__global__ __launch_bounds__(256)
void attn_kernel(const float* __restrict__ q, const float* __restrict__ k,
                 const float* __restrict__ v, float* __restrict__ out) {
  __shared__ float qs[64][64];
  __shared__ float ks[64][64];
  __shared__ float vs[64][64];

  const int b = blockIdx.x / N_;
  const int n = blockIdx.x % N_;
  const int t = threadIdx.x;

  const int r  = t >> 2;           // time row 0..63
  const int cb = (t & 3) * 16;     // column base
  const size_t base = ((size_t)(b * L_ + r) * N_ + n) * D_ + cb;
#pragma unroll
  for (int j = 0; j < 4; ++j) {
    *(v4f*)&qs[r][cb + 4 * j] = *(const v4f*)(q + base + 4 * j);
    *(v4f*)&ks[r][cb + 4 * j] = *(const v4f*)(k + base + 4 * j);
    *(v4f*)&vs[r][cb + 4 * j] = *(const v4f*)(v + base + 4 * j);
  }
  __syncthreads();

  const float scale = 0.35355339059327373f;  // 1/sqrt(8)
#pragma unroll
  for (int rr = 0; rr < 2; ++rr) {
    const int hr = t + rr * 256;   // 0..511
    const int h  = hr >> 6;
    const int l  = hr & 63;

    float qreg[DK_];
#pragma unroll
    for (int e = 0; e < DK_; ++e) qreg[e] = qs[l][h * DK_ + e];

    float p[64];
    float mx = -1e30f;
#pragma unroll
    for (int s = 0; s < 64; ++s) {
      float d = 0.f;
#pragma unroll
      for (int e = 0; e < DK_; ++e) d += qreg[e] * ks[s][h * DK_ + e];
      d *= scale;
      p[s] = d;
      mx = fmaxf(mx, d);
    }
    float sum = 0.f;
#pragma unroll
    for (int s = 0; s < 64; ++s) {
      p[s] = __expf(p[s] - mx);
      sum += p[s];
    }
    const float inv = 1.f / sum;
#pragma unroll
    for (int d = 0; d < DK_; ++d) {
      float a = 0.f;
#pragma unroll
      for (int s = 0; s < 64; ++s) a += p[s] * vs[s][h * DK_ + d];
      out[((size_t)(b * L_ + l) * N_ + n) * D_ + h * DK_ + d] = a * inv;
    }
  }
}

// ---------------------------------------------------------------------------
extern "C" void kernel_launch(void* const* d_in, const int* in_sizes, int n_in,
                              void* d_out, int out_size, void* d_ws, size_t ws_size,
                              hipStream_t stream) {
  const float* queries = (const float*)d_in[0];
  const float* keys    = (const float*)d_in[1];
  const float* values  = (const float*)d_in[2];
  const float* support = (const float*)d_in[3];   // [1, N, N]
  const float* Wq = (const float*)d_in[4];
  const float* bq = (const float*)d_in[5];
  const float* Wk = (const float*)d_in[6];
  const float* bk = (const float*)d_in[7];
  const float* Wv = (const float*)d_in[8];
  const float* bv = (const float*)d_in[9];
  const float* Wo = (const float*)d_in[10];
  const float* bo = (const float*)d_in[11];
  const float* Wg1 = (const float*)d_in[12];
  const float* bg1 = (const float*)d_in[13];
  const float* Wg2 = (const float*)d_in[14];
  const float* bg2 = (const float*)d_in[15];
  float* out = (float*)d_out;

  const size_t SLICE = (size_t)B_ * S_ * N_ * D_;  // 12,582,912 floats
  float* ws   = (float*)d_ws;
  float* W0   = ws;                 // hop1 / attn-out scratch
  float* W1   = ws + 1 * SLICE;     // hop2 scratch
  float* Kbuf = ws + 2 * SLICE;     // projected K
  float* Vbuf = ws + 3 * SLICE;     // projected V
  float* Qbuf = ws + 4 * SLICE;     // projected Q
  float* Wkg  = ws + 5 * SLICE;     // fused K-side weights [64,192]
  float* bkg  = Wkg + 64 * CIN_;
  float* Wvg  = bkg + 64;
  float* bvg  = Wvg + 64 * CIN_;

  // 1) Fuse mlp+projection weights (tiny).
  combine_w_kernel<<<1, 256, 0, stream>>>(Wk, bk, Wg1, bg1, Wkg, bkg);
  combine_w_kernel<<<1, 256, 0, stream>>>(Wv, bv, Wg2, bg2, Wvg, bvg);

  const dim3 hgrid(N_ / 64, B_ * S_);
  const int  prows = (B_ * S_ * N_) / 64;  // 3072 row-blocks

  // 2) Keys: two diffusion hops, then fused concat-projection -> K.
  hop_kernel<<<hgrid, 256, 0, stream>>>(support, keys, W0);
  hop_kernel<<<hgrid, 256, 0, stream>>>(support, W0, W1);
  proj_kernel<<<prows, 256, 0, stream>>>(keys, W0, W1, 3, Wkg, CIN_, bkg, Kbuf);

  // 3) Values: same pipeline -> V (reuses W0/W1).
  hop_kernel<<<hgrid, 256, 0, stream>>>(support, values, W0);
  hop_kernel<<<hgrid, 256, 0, stream>>>(support, W0, W1);
  proj_kernel<<<prows, 256, 0, stream>>>(values, W0, W1, 3, Wvg, CIN_, bvg, Vbuf);

  // 4) Q projection.
  proj_kernel<<<prows, 256, 0, stream>>>(queries, nullptr, nullptr, 1, Wq, 64, bq, Qbuf);

  // 5) Per-node attention over time -> W0.
  attn_kernel<<<B_ * N_, 256, 0, stream>>>(Qbuf, Kbuf, Vbuf, W0);

  // 6) Output projection -> d_out.
  proj_kernel<<<prows, 256, 0, stream>>>(W0, nullptr, nullptr, 1, Wo, 64, bo, out);
}